// FlexDecodingAttention_20452634263885
// MI455X (gfx1250) — compile-verified
//
#include <hip/hip_runtime.h>
#include <math.h>

// Problem constants (B=1, S=8192, H=16, D=128, DIM=2048, fp32).
#define DIM     2048
#define HN      16
#define HD      128
#define SEQ     8192
#define CHUNK   128                 // seq rows per wave in flash-decode split
#define NCHUNK  (SEQ / CHUNK)       // 64 chunks per head
#define PSTRIDE 132                 // floats per partial record: m, l, pad, pad, acc[128]
#define QK_SCALE 0.08838834764831845f   // 1/sqrt(128)
#define LOG2E    1.4426950408889634f

typedef __attribute__((ext_vector_type(2))) float v2f;
typedef __attribute__((ext_vector_type(8))) float v8f;
typedef __attribute__((ext_vector_type(4))) unsigned int u32x4;
typedef __attribute__((ext_vector_type(8))) int i32x8;
typedef __attribute__((ext_vector_type(4))) int i32x4;

// ---------------------------------------------------------------------------
// TDM: async 2D tile load Global -> LDS (rows x 128 f32, row stride 2048 f32).
// D# packed per CDNA5 ISA §8.3-8.5: group0 {count=1, lds_addr, global_addr,
// type=2}; group1 {data_size=4B, tensor_dim0=128, tensor_dim1=rows,
// tile_dim0=128, tile_dim1=rows, tensor_dim0_stride=2048}; groups 2/3 zero
// (2D tensor). Tracked by TENSORcnt. 6-arg builtin form (this toolchain).
// ---------------------------------------------------------------------------
__device__ __forceinline__ void tdm_load_v_tile(unsigned lds_byte,
                                                const float* gaddr,
                                                unsigned rows) {
  const unsigned long long ga = (unsigned long long)(const void*)gaddr;
  u32x4 g0;
  g0[0] = 1u;                                        // count=1, no gather
  g0[1] = lds_byte;                                  // LDS dest byte address
  g0[2] = (unsigned)(ga & 0xFFFFFFFFu);              // global_addr[31:0]
  g0[3] = (unsigned)((ga >> 32) & 0x1FFFFFFu)        // global_addr[56:32]
          | (2u << 30);                              // type=2 ("image")
  i32x8 g1;
  g1[0] = (int)(2u << 16);                           // wg_mask=0, data_size=2 (4B)
  g1[1] = (int)((128u & 0xFFFFu) << 16);             // tensor_dim0[15:0]=128
  g1[2] = (int)((rows & 0xFFFFu) << 16);             // dim0 hi=0 | tensor_dim1 lo
  g1[3] = (int)(128u << 16);                         // dim1 hi=0 | tile_dim0=128
  g1[4] = (int)(rows & 0xFFFFu);                     // tile_dim1=rows, tile_dim2=0
  g1[5] = (int)2048u;                                // tensor_dim0_stride = H*D
  g1[6] = 0;                                         // stride hi | dim1_stride lo
  g1[7] = 0;                                         // dim1_stride hi
  const i32x4 z4 = {0, 0, 0, 0};
  const i32x8 z8 = {0, 0, 0, 0, 0, 0, 0, 0};
  __builtin_amdgcn_tensor_load_to_lds(g0, g1, z4, z4, z8, 0);
}

// ---------------------------------------------------------------------------
// Kernel 1: fused q/k/v projections. out[i] = dot(x, W[i,:]) for W in {wq,wk,wv}.
// ---------------------------------------------------------------------------
__global__ void qkv_gemv(const float* __restrict__ x,
                         const float* __restrict__ wq,
                         const float* __restrict__ wk,
                         const float* __restrict__ wv,
                         float* __restrict__ qkv) {
  __shared__ float xs[DIM];
  for (int i = threadIdx.x; i < DIM; i += blockDim.x) xs[i] = x[i];
  __syncthreads();

  const int lane = threadIdx.x & 31;
  const int gw   = blockIdx.x * (blockDim.x >> 5) + (threadIdx.x >> 5);
  const int which = gw >> 11;          // 0=q, 1=k, 2=v
  const int row   = gw & (DIM - 1);
  if (which >= 3) return;
  const float* W  = (which == 0) ? wq : (which == 1) ? wk : wv;
  const float* wr = W + (size_t)row * DIM;

  float acc = 0.f;
#pragma unroll
  for (int t = 0; t < DIM / 128; ++t) {
    const int j = t * 128 + lane * 4;
    const float4 w4 = *(const float4*)(wr + j);
    const float4 x4 = *(const float4*)(xs + j);
    acc += w4.x * x4.x + w4.y * x4.y + w4.z * x4.z + w4.w * x4.w;
  }
  for (int off = 16; off > 0; off >>= 1) acc += __shfl_xor(acc, off, 32);
  if (lane == 0) qkv[which * DIM + row] = acc;
}

// ---------------------------------------------------------------------------
// Kernel 2: flash-decoding partial attention. One wave32 per (head, 128-row chunk).
//
// Scores: chained V_WMMA_F32_16X16X4_F32. Contraction step reordering makes
// each lane's A fragment a contiguous 256B span of its K row:
//   step 2i   contracts d = half*64 + 4i + {0,1}
//   step 2i+1 contracts d = half*64 + 4i + {2,3}
// (A and B use the identical (half,slot)->d map; B replicates q across all 16
// columns so every D column holds dot(K_row, q).)  K fragment = 16 batched
// global_load_b128 -> one wait -> 32 WMMAs (real memory-level parallelism).
//
// V: double-buffered TDM prefetch into LDS (tile t+1 in flight while tile t
// computes), s_wait_tensorcnt(1) before consuming (TDM completes in order).
// Row s == position is patched from k_new/v_new; inputs never mutated.
// ---------------------------------------------------------------------------
__global__ void attn_partial(const float* __restrict__ kc,
                             const float* __restrict__ vc,
                             const float* __restrict__ qkv,
                             float* __restrict__ part,
                             const int* __restrict__ posp) {
  __shared__ float vbuf[2][16 * HD];       // 2 x 8KB V tiles

  const int lane  = threadIdx.x;           // wave32
  const int chunk = blockIdx.x;            // 0..NCHUNK-1
  const int h     = blockIdx.y;            // head
  const int position = *posp;
  const int m    = lane & 15;
  const int half = lane >> 4;

  const float* qh   = qkv + h * HD;
  const float* knew = qkv + DIM + h * HD;
  const float* vnew = qkv + 2 * DIM + h * HD;

  // q fragment: this lane's contiguous 64 floats (d = half*64 .. half*64+63).
  float4 qf[16];
#pragma unroll
  for (int i = 0; i < 16; ++i)
    qf[i] = *(const float4*)(qh + half * 64 + i * 4);

  const int s_base = chunk * CHUNK;

  // Kick off TDM for tile 0's V.
  tdm_load_v_tile((unsigned)(unsigned long long)(const void*)&vbuf[0][0],
                  vc + ((size_t)s_base * HN + h) * HD, 16u);

  float  m_run = -INFINITY;
  float  l_run = 0.f;
  float4 acc   = make_float4(0.f, 0.f, 0.f, 0.f);

  for (int tile = 0; tile < CHUNK / 16; ++tile) {
    const int s0 = s_base + tile * 16;

    // Prefetch next tile's V while this tile computes.
    if (tile < CHUNK / 16 - 1) {
      tdm_load_v_tile(
          (unsigned)(unsigned long long)(const void*)&vbuf[(tile + 1) & 1][0],
          vc + ((size_t)(s0 + 16) * HN + h) * HD, 16u);
    }

    // --- K fragment: 16 batched B128 loads (contiguous 256B per lane). ---
    const int srow = s0 + m;
    const float* krow = ((srow == position) ? knew
                                            : (kc + ((size_t)srow * HN + h) * HD))
                        + half * 64;
    float4 kf[16];
#pragma unroll
    for (int i = 0; i < 16; ++i) kf[i] = *(const float4*)(krow + i * 4);

    // --- 32 chained fp32 WMMAs: D[r][*] = dot(K_row, q). ---
    v8f d8 = {};
#pragma unroll
    for (int i = 0; i < 16; ++i) {
      v2f a, b;
      a.x = kf[i].x; a.y = kf[i].y;
      b.x = qf[i].x; b.y = qf[i].y;
      d8 = __builtin_amdgcn_wmma_f32_16x16x4_f32(false, a, false, b,
                                                 (short)0, d8, false, false);
      a.x = kf[i].z; a.y = kf[i].w;
      b.x = qf[i].z; b.y = qf[i].w;
      d8 = __builtin_amdgcn_wmma_f32_16x16x4_f32(false, a, false, b,
                                                 (short)0, d8, false, false);
    }

    // Scale, mask rows beyond position, tile max.
    // C/D layout: lanes 0-15 VGPR r -> row s0+r ; lanes 16-31 -> row s0+8+r.
    float sc[8];
    float tmax = -INFINITY;
#pragma unroll
    for (int r = 0; r < 8; ++r) {
      const int rrow = s0 + r + half * 8;
      const float v  = d8[r] * QK_SCALE;
      sc[r] = (rrow <= position) ? v : -INFINITY;
      tmax  = fmaxf(tmax, sc[r]);
    }
    tmax = fmaxf(tmax, __shfl_xor(tmax, 16, 32));
    const float m_new = fmaxf(m_run, tmax);
    if (m_new == -INFINITY) continue;            // whole tile masked (uniform)

    const float alpha = (m_run == -INFINITY) ? 0.f : exp2f((m_run - m_new) * LOG2E);
    acc.x *= alpha; acc.y *= alpha; acc.z *= alpha; acc.w *= alpha;
    l_run *= alpha;
    m_run  = m_new;

    // --- Consume this tile's V from LDS (TDM in-order: <=1 outstanding). ---
    if (tile < CHUNK / 16 - 1) __builtin_amdgcn_s_wait_tensorcnt(1);
    else                       __builtin_amdgcn_s_wait_tensorcnt(0);

    float* vt = &vbuf[tile & 1][0];
    if (s0 <= position && position < s0 + 16) {  // patch new-token row (uniform)
      *(float4*)(vt + (position - s0) * HD + lane * 4) =
          *(const float4*)(vnew + lane * 4);
    }

#pragma unroll
    for (int r = 0; r < 16; ++r) {
      const float s_r = __shfl(sc[r & 7], (r < 8) ? 0 : 16, 32);
      const float p   = ((s0 + r) <= position) ? exp2f((s_r - m_new) * LOG2E) : 0.f;
      l_run += p;
      const float4 v4 = *(const float4*)(vt + r * HD + lane * 4);
      acc.x += p * v4.x; acc.y += p * v4.y; acc.z += p * v4.z; acc.w += p * v4.w;
    }
  }

  float* pr = part + ((size_t)h * NCHUNK + chunk) * PSTRIDE;
  if (lane == 0) { pr[0] = m_run; pr[1] = l_run; }
  *(float4*)(pr + 4 + lane * 4) = acc;
}

// ---------------------------------------------------------------------------
// Kernel 3: log-sum-exp merge of NCHUNK partials per head -> o[h][d].
// ---------------------------------------------------------------------------
__global__ void attn_combine(const float* __restrict__ part,
                             float* __restrict__ o) {
  const int h = blockIdx.x;
  const int d = threadIdx.x;           // 0..127
  const float* ph = part + (size_t)h * NCHUNK * PSTRIDE;
  float gmax = -INFINITY;
  for (int c = 0; c < NCHUNK; ++c) gmax = fmaxf(gmax, ph[c * PSTRIDE]);
  float l_tot = 0.f, num = 0.f;
  for (int c = 0; c < NCHUNK; ++c) {
    const float mc = ph[c * PSTRIDE];
    const float w  = (mc == -INFINITY) ? 0.f : exp2f((mc - gmax) * LOG2E);
    l_tot += w * ph[c * PSTRIDE + 1];
    num   += w * ph[c * PSTRIDE + 4 + d];
  }
  o[h * HD + d] = num / l_tot;
}

// ---------------------------------------------------------------------------
// Kernel 4: final projection out[i] = dot(o_flat, w_o[i,:]).
// ---------------------------------------------------------------------------
__global__ void out_gemv(const float* __restrict__ o,
                         const float* __restrict__ wo,
                         float* __restrict__ out) {
  __shared__ float xs[DIM];
  for (int i = threadIdx.x; i < DIM; i += blockDim.x) xs[i] = o[i];
  __syncthreads();

  const int lane = threadIdx.x & 31;
  const int row  = blockIdx.x * (blockDim.x >> 5) + (threadIdx.x >> 5);
  const float* wr = wo + (size_t)row * DIM;

  float acc = 0.f;
#pragma unroll
  for (int t = 0; t < DIM / 128; ++t) {
    const int j = t * 128 + lane * 4;
    const float4 w4 = *(const float4*)(wr + j);
    const float4 x4 = *(const float4*)(xs + j);
    acc += w4.x * x4.x + w4.y * x4.y + w4.z * x4.z + w4.w * x4.w;
  }
  for (int off = 16; off > 0; off >>= 1) acc += __shfl_xor(acc, off, 32);
  if (lane == 0) out[row] = acc;
}

// ---------------------------------------------------------------------------
extern "C" void kernel_launch(void* const* d_in, const int* in_sizes, int n_in,
                              void* d_out, int out_size, void* d_ws, size_t ws_size,
                              hipStream_t stream) {
  const float* x  = (const float*)d_in[0];
  const float* kc = (const float*)d_in[1];
  const float* vc = (const float*)d_in[2];
  const float* wq = (const float*)d_in[3];
  const float* wk = (const float*)d_in[4];
  const float* wv = (const float*)d_in[5];
  const float* wo = (const float*)d_in[6];
  const int*  pos = (const int*)d_in[7];

  float* ws   = (float*)d_ws;
  float* qkv  = ws;                          // q | k_new | v_new : 3*DIM floats
  float* ov   = ws + 3 * DIM;                // attention output  : DIM floats
  float* part = ws + 4 * DIM;                // HN*NCHUNK*PSTRIDE floats (~528 KB)

  qkv_gemv<<<(3 * DIM) / 8, 256, 0, stream>>>(x, wq, wk, wv, qkv);
  attn_partial<<<dim3(NCHUNK, HN), 32, 0, stream>>>(kc, vc, qkv, part, pos);
  attn_combine<<<HN, HD, 0, stream>>>(part, ov);
  out_gemv<<<DIM / 8, 256, 0, stream>>>(ov, wo, (float*)d_out);
}